// Conf_aware_Label_Gen_module_29472065585162
// MI455X (gfx1250) — compile-verified
//
#include <hip/hip_runtime.h>
#include <math.h>

// ---------------- problem constants (match reference) ----------------
#define BATCH 2
#define NPTS  8192
#define NVOX  64
#define SEGS  (BATCH * NVOX)        // 128
#define MIN_PTS_C 20
#define EPS_C   1e-6f
#define THETA0_C 0.01f
#define FLOW_THR_C 0.2f
#define POS_THR_C  0.1f

typedef __attribute__((ext_vector_type(2))) float v2f;
typedef __attribute__((ext_vector_type(8))) float v8f;

// =====================================================================
// KNN via V_WMMA_F32_16X16X4_F32
//   A (16x4, refs):    lane<16: (x, y)     lane>=16: (z, |r|^2)
//   B (4x16, queries): lane<16: (-2qx,-2qy) lane>=16: (-2qz, 1)
//   D[M=ref][N=query] = |r|^2 - 2 q.r   -> argmin over refs
// 4 waves/block; each wave owns TWO 16-query tiles (2 WMMAs per A tile).
// Argmin: min-reduce the 8 accumulators, then a branch-guarded index
// recovery that is skipped (EXECZ) for almost every tile.
// =====================================================================
#define KNN_WAVES 4
#define QPW 32
#define CHUNK 256

__global__ void __launch_bounds__(128)
knn_wmma_kernel(const float* __restrict__ pc1, const float* __restrict__ pg,
                const float* __restrict__ pc2, int* __restrict__ out_idx)
{
    __shared__ float4 sref[CHUNK];
    const int b    = blockIdx.y;
    const int wave = threadIdx.x >> 5;
    const int lane = threadIdx.x & 31;
    const int col  = lane & 15;
    const int hi   = (lane >= 16) ? 1 : 0;
    const int qbase = (blockIdx.x * KNN_WAVES + wave) * QPW;

    const float* p1  = pc1 + (size_t)b * NPTS * 3;
    const float* pgb = pg  + (size_t)b * NPTS * 3;
    const float* rf  = pc2 + (size_t)b * NPTS * 3;

    // Two B operands (query groups) -- constant across the whole scan
    v2f bm0, bm1;
    {
        int qi = qbase + col;
        float qx = p1[3 * qi + 0] + pgb[3 * qi + 0];
        float qy = p1[3 * qi + 1] + pgb[3 * qi + 1];
        float qz = p1[3 * qi + 2] + pgb[3 * qi + 2];
        bm0.x = hi ? (-2.0f * qz) : (-2.0f * qx);
        bm0.y = hi ? 1.0f         : (-2.0f * qy);
        qi += 16;
        qx = p1[3 * qi + 0] + pgb[3 * qi + 0];
        qy = p1[3 * qi + 1] + pgb[3 * qi + 1];
        qz = p1[3 * qi + 2] + pgb[3 * qi + 2];
        bm1.x = hi ? (-2.0f * qz) : (-2.0f * qx);
        bm1.y = hi ? 1.0f         : (-2.0f * qy);
    }

    float best0 = 3.402823466e38f, best1 = 3.402823466e38f;
    int   bi0 = 0, bi1 = 0;

    for (int cb = 0; cb < NPTS; cb += CHUNK) {
        __syncthreads();
        for (int r = threadIdx.x; r < CHUNK; r += 128) {
            const float x = rf[3 * (cb + r) + 0];
            const float y = rf[3 * (cb + r) + 1];
            const float z = rf[3 * (cb + r) + 2];
            sref[r] = make_float4(x, y, z, x * x + y * y + z * z);
        }
        __syncthreads();

        for (int t = 0; t < CHUNK / 16; ++t) {
            const float4 rv = sref[t * 16 + col];
            v2f am;
            am.x = hi ? rv.z : rv.x;
            am.y = hi ? rv.w : rv.y;
            const v8f z8 = {0.f, 0.f, 0.f, 0.f, 0.f, 0.f, 0.f, 0.f};
            v8f a0 = __builtin_amdgcn_wmma_f32_16x16x4_f32(
                false, am, false, bm0, (short)0, z8, false, false);
            v8f a1 = __builtin_amdgcn_wmma_f32_16x16x4_f32(
                false, am, false, bm1, (short)0, z8, false, false);
            const int rowoff = cb + t * 16 + (hi ? 8 : 0);

            // group 0: cheap min-reduce, rare index recovery
            const float t0 = fminf(fminf(fminf(a0[0], a0[1]), fminf(a0[2], a0[3])),
                                   fminf(fminf(a0[4], a0[5]), fminf(a0[6], a0[7])));
            if (t0 < best0) {
                best0 = t0;
                int sel = rowoff + 7;
#pragma unroll
                for (int m = 6; m >= 0; --m) sel = (a0[m] == t0) ? (rowoff + m) : sel;
                bi0 = sel;
            }
            // group 1
            const float t1 = fminf(fminf(fminf(a1[0], a1[1]), fminf(a1[2], a1[3])),
                                   fminf(fminf(a1[4], a1[5]), fminf(a1[6], a1[7])));
            if (t1 < best1) {
                best1 = t1;
                int sel = rowoff + 7;
#pragma unroll
                for (int m = 6; m >= 0; --m) sel = (a1[m] == t1) ? (rowoff + m) : sel;
                bi1 = sel;
            }
        }
    }
    // merge the two row-halves (lane L and L+16 track the same query column)
    {
        const float ob = __shfl_xor(best0, 16, 32);
        const int   oi = __shfl_xor(bi0, 16, 32);
        if (ob < best0 || (ob == best0 && oi < bi0)) { best0 = ob; bi0 = oi; }
    }
    {
        const float ob = __shfl_xor(best1, 16, 32);
        const int   oi = __shfl_xor(bi1, 16, 32);
        if (ob < best1 || (ob == best1 && oi < bi1)) { best1 = ob; bi1 = oi; }
    }
    if (lane < 16) {
        out_idx[(size_t)b * NPTS + qbase + lane]      = bi0;
        out_idx[(size_t)b * NPTS + qbase + 16 + lane] = bi1;
    }
}

// =====================================================================
// Voxel population counts -> voxel_big flags (one small block)
// =====================================================================
__global__ void counts_kernel(const int* __restrict__ vox, int* __restrict__ voxel_big)
{
    __shared__ int cnt[SEGS];
    const int t = threadIdx.x;
    for (int i = t; i < SEGS; i += 256) cnt[i] = 0;
    __syncthreads();
    for (int i = t; i < BATCH * NPTS; i += 256) {
        const int b = i / NPTS;
        atomicAdd(&cnt[b * NVOX + vox[i]], 1);
    }
    __syncthreads();
    for (int i = t; i < SEGS; i += 256) voxel_big[i] = (cnt[i] > MIN_PTS_C) ? 1 : 0;
}

// =====================================================================
// Per-point confidence weights + gathered nn_pc2
// =====================================================================
__global__ void weights_kernel(const float* __restrict__ pc1, const float* __restrict__ pg,
                               const float* __restrict__ pc2,
                               const float* __restrict__ ffwd, const float* __restrict__ fbwd,
                               const int* __restrict__ idx, const int* __restrict__ conf_flag,
                               float* __restrict__ w, float* __restrict__ nn2out)
{
    const int i = blockIdx.x * blockDim.x + threadIdx.x;
    if (i >= BATCH * NPTS) return;
    const int b = i / NPTS;
    const int j = idx[i];
    const float* p2 = pc2 + ((size_t)b * NPTS + j) * 3;
    const float nx = p2[0], ny = p2[1], nz = p2[2];
    nn2out[3 * i + 0] = nx; nn2out[3 * i + 1] = ny; nn2out[3 * i + 2] = nz;

    if (*conf_flag) {
        const float* fb = fbwd + ((size_t)b * NPTS + j) * 3;
        const float f0 = ffwd[3 * i + 0] + fb[0];
        const float f1 = ffwd[3 * i + 1] + fb[1];
        const float f2 = ffwd[3 * i + 2] + fb[2];
        const float flow_diff = f0 * f0 + f1 * f1 + f2 * f2 + EPS_C;
        const float d0 = pc1[3 * i + 0] + pg[3 * i + 0] - nx;
        const float d1 = pc1[3 * i + 1] + pg[3 * i + 1] - ny;
        const float d2 = pc1[3 * i + 2] + pg[3 * i + 2] - nz;
        const float pos_diff = sqrtf(d0 * d0 + d1 * d1 + d2 * d2 + EPS_C);
        const float conf = expf(-flow_diff / THETA0_C) + EPS_C;
        const float g = ((flow_diff < FLOW_THR_C) ? 1.f : 0.f) *
                        ((pos_diff  < POS_THR_C)  ? 1.f : 0.f);
        w[i] = conf * g + EPS_C;
    } else {
        w[i] = 1.0f + EPS_C;
    }
}

// =====================================================================
// Segment sums: per voxel [w, w*x(3), w*y(3), w*x_i*y_j(9)] = 16 floats
// LDS-accumulated (ds_add_f32) then flushed with global atomics.
// =====================================================================
__global__ void zero_kernel(float* __restrict__ p, int n)
{
    const int i = blockIdx.x * blockDim.x + threadIdx.x;
    if (i < n) p[i] = 0.0f;
}

__global__ void accum_kernel(const float* __restrict__ pc1, const float* __restrict__ nn2,
                             const float* __restrict__ w, const int* __restrict__ vox,
                             float* __restrict__ gsum)
{
    __shared__ float acc[NVOX * 16];
    const int b = blockIdx.y;
    const int t = threadIdx.x;
    for (int i = t; i < NVOX * 16; i += 256) acc[i] = 0.0f;
    __syncthreads();

    const int p  = blockIdx.x * 256 + t;
    const int gi = b * NPTS + p;
    const int v  = vox[gi];
    const float ww = w[gi];
    const float x0 = pc1[3 * gi + 0], x1 = pc1[3 * gi + 1], x2 = pc1[3 * gi + 2];
    const float y0 = nn2[3 * gi + 0], y1 = nn2[3 * gi + 1], y2 = nn2[3 * gi + 2];
    float* a = &acc[v * 16];
    atomicAdd(a + 0, ww);
    atomicAdd(a + 1, ww * x0); atomicAdd(a + 2, ww * x1); atomicAdd(a + 3, ww * x2);
    atomicAdd(a + 4, ww * y0); atomicAdd(a + 5, ww * y1); atomicAdd(a + 6, ww * y2);
    const float xs[3] = {x0, x1, x2};
    const float ys[3] = {y0, y1, y2};
#pragma unroll
    for (int ii = 0; ii < 3; ++ii)
#pragma unroll
        for (int jj = 0; jj < 3; ++jj)
            atomicAdd(a + 7 + 3 * ii + jj, ww * xs[ii] * ys[jj]);
    __syncthreads();

    float* g = gsum + (size_t)b * NVOX * 16;
    for (int i = t; i < NVOX * 16; i += 256) atomicAdd(&g[i], acc[i]);
}

// =====================================================================
// Per-segment weighted Kabsch: 3x3 Jacobi eigen of A^T A -> V, rebuild U,
// R = V diag(1,1,det) U^T,  t = m2 - R m1.  One thread per segment.
// =====================================================================
__global__ void kabsch_kernel(const float* __restrict__ gsum,
                              float* __restrict__ Rb, float* __restrict__ tb)
{
    const int s = threadIdx.x;
    if (s >= SEGS) return;
    const float* q = gsum + s * 16;
    const float denom = fmaxf(q[0], 1e-12f);
    float m1[3], m2[3];
#pragma unroll
    for (int k = 0; k < 3; ++k) { m1[k] = q[1 + k] / denom; m2[k] = q[4 + k] / denom; }
    float A[3][3];
#pragma unroll
    for (int i = 0; i < 3; ++i)
#pragma unroll
        for (int j = 0; j < 3; ++j)
            A[i][j] = q[7 + 3 * i + j] - denom * m1[i] * m2[j];

    // S = A^T A (symmetric PSD)
    float S[3][3];
#pragma unroll
    for (int i = 0; i < 3; ++i)
#pragma unroll
        for (int j = 0; j < 3; ++j)
            S[i][j] = A[0][i] * A[0][j] + A[1][i] * A[1][j] + A[2][i] * A[2][j];

    float V[3][3] = {{1, 0, 0}, {0, 1, 0}, {0, 0, 1}};
    const int PP[3] = {0, 0, 1}, QQ[3] = {1, 2, 2};
    for (int sweep = 0; sweep < 10; ++sweep) {
        for (int r = 0; r < 3; ++r) {
            const int p = PP[r], qd = QQ[r];
            const float apq = S[p][qd];
            if (fabsf(apq) < 1e-14f) continue;
            const float tau = (S[qd][qd] - S[p][p]) / (2.0f * apq);
            const float tt = ((tau >= 0.f) ? 1.f : -1.f) / (fabsf(tau) + sqrtf(1.f + tau * tau));
            const float c = 1.0f / sqrtf(1.0f + tt * tt);
            const float sn = tt * c;
            for (int k = 0; k < 3; ++k) {
                const float skp = S[k][p], skq = S[k][qd];
                S[k][p] = c * skp - sn * skq; S[k][qd] = sn * skp + c * skq;
            }
            for (int k = 0; k < 3; ++k) {
                const float spk = S[p][k], sqk = S[qd][k];
                S[p][k] = c * spk - sn * sqk; S[qd][k] = sn * spk + c * sqk;
            }
            for (int k = 0; k < 3; ++k) {
                const float vkp = V[k][p], vkq = V[k][qd];
                V[k][p] = c * vkp - sn * vkq; V[k][qd] = sn * vkp + c * vkq;
            }
        }
    }
    float eig[3] = {S[0][0], S[1][1], S[2][2]};
    for (int i = 0; i < 2; ++i)
        for (int j = 0; j < 2 - i; ++j)
            if (eig[j] < eig[j + 1]) {
                const float e = eig[j]; eig[j] = eig[j + 1]; eig[j + 1] = e;
                for (int k = 0; k < 3; ++k) {
                    const float v0 = V[k][j]; V[k][j] = V[k][j + 1]; V[k][j + 1] = v0;
                }
            }

    // U columns from A * v_k, orthonormalized
    float u0[3], u1[3], u2[3];
#pragma unroll
    for (int i = 0; i < 3; ++i)
        u0[i] = A[i][0] * V[0][0] + A[i][1] * V[1][0] + A[i][2] * V[2][0];
    float n0 = sqrtf(u0[0] * u0[0] + u0[1] * u0[1] + u0[2] * u0[2]);
    if (n0 > 1e-9f) { u0[0] /= n0; u0[1] /= n0; u0[2] /= n0; }
    else { u0[0] = 1.f; u0[1] = 0.f; u0[2] = 0.f; }
#pragma unroll
    for (int i = 0; i < 3; ++i)
        u1[i] = A[i][0] * V[0][1] + A[i][1] * V[1][1] + A[i][2] * V[2][1];
    const float d01 = u1[0] * u0[0] + u1[1] * u0[1] + u1[2] * u0[2];
#pragma unroll
    for (int i = 0; i < 3; ++i) u1[i] -= d01 * u0[i];
    float n1 = sqrtf(u1[0] * u1[0] + u1[1] * u1[1] + u1[2] * u1[2]);
    if (n1 > 1e-9f) { u1[0] /= n1; u1[1] /= n1; u1[2] /= n1; }
    else {
        float tv[3] = {1.f, 0.f, 0.f};
        if (fabsf(u0[0]) > 0.9f) { tv[0] = 0.f; tv[1] = 1.f; }
        u1[0] = u0[1] * tv[2] - u0[2] * tv[1];
        u1[1] = u0[2] * tv[0] - u0[0] * tv[2];
        u1[2] = u0[0] * tv[1] - u0[1] * tv[0];
        const float nn = sqrtf(u1[0] * u1[0] + u1[1] * u1[1] + u1[2] * u1[2]) + 1e-20f;
        u1[0] /= nn; u1[1] /= nn; u1[2] /= nn;
    }
    u2[0] = u0[1] * u1[2] - u0[2] * u1[1];
    u2[1] = u0[2] * u1[0] - u0[0] * u1[2];
    u2[2] = u0[0] * u1[1] - u0[1] * u1[0];

    const float detV = V[0][0] * (V[1][1] * V[2][2] - V[1][2] * V[2][1])
                     - V[0][1] * (V[1][0] * V[2][2] - V[1][2] * V[2][0])
                     + V[0][2] * (V[1][0] * V[2][1] - V[1][1] * V[2][0]);
    const float detR = detV;  // det(U) == +1 by construction

    float R[9];
#pragma unroll
    for (int i = 0; i < 3; ++i)
#pragma unroll
        for (int j = 0; j < 3; ++j)
            R[3 * i + j] = V[i][0] * u0[j] + V[i][1] * u1[j] + detR * V[i][2] * u2[j];

#pragma unroll
    for (int i = 0; i < 9; ++i) Rb[s * 9 + i] = R[i];
#pragma unroll
    for (int i = 0; i < 3; ++i)
        tb[s * 3 + i] = m2[i] - (R[3 * i + 0] * m1[0] + R[3 * i + 1] * m1[1] + R[3 * i + 2] * m1[2]);
}

// =====================================================================
// Apply rigid fit to points in big voxels: pseudo_gt = R pc1 + t - pc1
// =====================================================================
__global__ void apply_kernel(const float* __restrict__ pc1, const int* __restrict__ vox,
                             const int* __restrict__ vb,
                             const float* __restrict__ Rb, const float* __restrict__ tb,
                             float* __restrict__ pg)
{
    const int i = blockIdx.x * blockDim.x + threadIdx.x;
    if (i >= BATCH * NPTS) return;
    const int b = i / NPTS;
    const int s = b * NVOX + vox[i];
    if (!vb[s]) return;
    const float* R = Rb + s * 9;
    const float* t = tb + s * 3;
    const float x = pc1[3 * i + 0], y = pc1[3 * i + 1], z = pc1[3 * i + 2];
    pg[3 * i + 0] = R[0] * x + R[1] * y + R[2] * z + t[0] - x;
    pg[3 * i + 1] = R[3] * x + R[4] * y + R[5] * z + t[1] - y;
    pg[3 * i + 2] = R[6] * x + R[7] * y + R[8] * z + t[2] - z;
}

// =====================================================================
// Final outputs: pseudo_gt | validity | pos_diff | flow_diff
// =====================================================================
__global__ void final_kernel(const float* __restrict__ pc1, const float* __restrict__ pg,
                             const float* __restrict__ pc2,
                             const float* __restrict__ ffwd, const float* __restrict__ fbwd,
                             const int* __restrict__ vox, const int* __restrict__ vb,
                             const int* __restrict__ idx, float* __restrict__ out)
{
    const int i = blockIdx.x * blockDim.x + threadIdx.x;
    if (i >= BATCH * NPTS) return;
    const int b = i / NPTS;
    const int j = idx[i];
    const float* p2 = pc2 + ((size_t)b * NPTS + j) * 3;
    const float* fb = fbwd + ((size_t)b * NPTS + j) * 3;

    const float g0 = pg[3 * i + 0], g1 = pg[3 * i + 1], g2 = pg[3 * i + 2];
    const float w0 = pc1[3 * i + 0] + g0 - p2[0];
    const float w1 = pc1[3 * i + 1] + g1 - p2[1];
    const float w2 = pc1[3 * i + 2] + g2 - p2[2];
    const float pos_diff = sqrtf(w0 * w0 + w1 * w1 + w2 * w2 + EPS_C);
    const float f0 = ffwd[3 * i + 0] + fb[0];
    const float f1 = ffwd[3 * i + 1] + fb[1];
    const float f2 = ffwd[3 * i + 2] + fb[2];
    const float flow_diff = sqrtf(f0 * f0 + f1 * f1 + f2 * f2 + EPS_C + EPS_C);

    const int BN = BATCH * NPTS;
    out[3 * i + 0] = g0; out[3 * i + 1] = g1; out[3 * i + 2] = g2;
    out[3 * BN + i]          = vb[b * NVOX + vox[i]] ? 1.0f : 0.0f;
    out[3 * BN + BN + i]     = pos_diff;
    out[3 * BN + 2 * BN + i] = flow_diff;
}

// =====================================================================
// Host-side launch sequence (graph-capture safe: only kernels + async d2d)
// =====================================================================
extern "C" void kernel_launch(void* const* d_in, const int* in_sizes, int n_in,
                              void* d_out, int out_size, void* d_ws, size_t ws_size,
                              hipStream_t stream)
{
    const float* pc1   = (const float*)d_in[0];
    const float* flow  = (const float*)d_in[1];
    const float* bflow = (const float*)d_in[2];
    const int*   vox   = (const int*)d_in[3];
    const float* pc2   = (const float*)d_in[4];
    const int*   conf  = (const int*)d_in[5];
    float* out = (float*)d_out;

    // workspace carve-up (floats then ints)
    float* pg   = (float*)d_ws;            // B*N*3
    float* nn2  = pg   + BATCH * NPTS * 3; // B*N*3
    float* wbuf = nn2  + BATCH * NPTS * 3; // B*N
    float* gsum = wbuf + BATCH * NPTS;     // SEGS*16
    float* Rb   = gsum + SEGS * 16;        // SEGS*9
    float* tb   = Rb   + SEGS * 9;         // SEGS*3
    int*   idxb = (int*)(tb + SEGS * 3);   // B*N
    int*   vb   = idxb + BATCH * NPTS;     // SEGS

    const int BN = BATCH * NPTS;
    const dim3 knn_grid(NPTS / (KNN_WAVES * QPW), BATCH);
    const dim3 acc_grid(NPTS / 256, BATCH);

    // pseudo_gt = flow_pred
    hipMemcpyAsync(pg, flow, (size_t)BN * 3 * sizeof(float),
                   hipMemcpyDeviceToDevice, stream);
    // voxel population flags (seg labels are constant across iterations)
    counts_kernel<<<1, 256, 0, stream>>>(vox, vb);

    for (int it = 0; it < 2; ++it) {
        knn_wmma_kernel<<<knn_grid, 128, 0, stream>>>(pc1, pg, pc2, idxb);
        weights_kernel<<<(BN + 255) / 256, 256, 0, stream>>>(
            pc1, pg, pc2, flow, bflow, idxb, conf, wbuf, nn2);
        zero_kernel<<<(SEGS * 16 + 255) / 256, 256, 0, stream>>>(gsum, SEGS * 16);
        accum_kernel<<<acc_grid, 256, 0, stream>>>(pc1, nn2, wbuf, vox, gsum);
        kabsch_kernel<<<1, 128, 0, stream>>>(gsum, Rb, tb);
        apply_kernel<<<(BN + 255) / 256, 256, 0, stream>>>(pc1, vox, vb, Rb, tb, pg);
    }

    knn_wmma_kernel<<<knn_grid, 128, 0, stream>>>(pc1, pg, pc2, idxb);
    final_kernel<<<(BN + 255) / 256, 256, 0, stream>>>(
        pc1, pg, pc2, flow, bflow, vox, vb, idxb, out);
    (void)in_sizes; (void)n_in; (void)out_size; (void)ws_size;
}